// GeomLossBarycentricPooling_25580825215417
// MI455X (gfx1250) — compile-verified
//
#include <hip/hip_runtime.h>

// ---------------------------------------------------------------------------
// GeomLoss barycentric pooling: per-graph entropic OT histogram (Sinkhorn).
// One workgroup per graph.  Cost matrix computed with v_wmma_f32_16x16x4_f32,
// stored pre-scaled as C' = -C/eps in LDS, and kept resident for all 50
// log-domain Sinkhorn iterations (zero HBM traffic in the iteration loop).
// All potentials are kept pre-scaled (f' = f/eps, g' = g/eps) so the hot
// loop is a single add + max / add + exp per element.
// ---------------------------------------------------------------------------

#define B_GRAPHS 512
#define KCB      32          // codebook size
#define DIM      64          // hidden dim
#define SAMP     4           // samples per node
#define MAXS     1024        // max samples per graph kept in LDS (mean ~781)
#define CPAD     36          // padded row stride of C (16B-aligned rows)
#define CBPAD    68          // padded row stride of codebook in LDS
#define INV_EPS  10.0f       // 1/eps, eps = 0.1
#define ITERS    50
#define NTHREADS 256
#define NWAVES   (NTHREADS / 32)
#define LOG_K    3.46573590279972654f   // log(32)

typedef __attribute__((ext_vector_type(2))) float v2f;
typedef __attribute__((ext_vector_type(8))) float v8f;

// dynamic LDS layout (floats)
#define OFF_C   0                             // C' = -C/eps, [MAXS][CPAD]
#define OFF_F   (MAXS * CPAD)                 // f' (aliased as xnorm before use)
#define OFF_CB  (OFF_F + MAXS)                // codebook, padded
#define OFF_CN  (OFF_CB + KCB * CBPAD)        // ||c_k||^2
#define OFF_G   (OFF_CN + KCB)                // g'
#define OFF_R1  (OFF_G + KCB)                 // 8x32 reduction (max)
#define OFF_R2  (OFF_R1 + NWAVES * KCB)       // 8x32 reduction (sum)
#define SMEM_FLOATS (OFF_R2 + NWAVES * KCB)   // 40640 floats = 162560 B

__global__ __launch_bounds__(NTHREADS, 1)
void ot_sinkhorn_hist_kernel(const float* __restrict__ X,        // [Nnodes*4, 64]
                             const float* __restrict__ codebook, // [32, 64]
                             const int*   __restrict__ batch_idx,// [Nnodes], sorted
                             float*       __restrict__ out,      // [512, 32]
                             int Nnodes)
{
    extern __shared__ __align__(16) float smem[];
    float* Cs  = smem + OFF_C;
    float* fS  = smem + OFF_F;
    float* xn  = smem + OFF_F;                // alias: dead before f is live
    float* cb  = smem + OFF_CB;
    float* cn  = smem + OFF_CN;
    float* gS  = smem + OFF_G;
    float* red = smem + OFF_R1;
    float* red2= smem + OFF_R2;
    __shared__ int sbounds[2];

    const int b    = blockIdx.x;
    const int tid  = threadIdx.x;
    const int lane = tid & 31;
    const int wv   = tid >> 5;

    // ---- segment bounds via binary search on sorted batch_idx -------------
    if (tid == 0) {
        int lo = 0, hi = Nnodes;
        while (lo < hi) { int mid = (lo + hi) >> 1; if (batch_idx[mid] < b) lo = mid + 1; else hi = mid; }
        sbounds[0] = lo;
        int lo2 = lo, hi2 = Nnodes;
        while (lo2 < hi2) { int mid = (lo2 + hi2) >> 1; if (batch_idx[mid] < b + 1) lo2 = mid + 1; else hi2 = mid; }
        sbounds[1] = lo2;
    }
    __syncthreads();
    const int node_start = sbounds[0];
    int n = (sbounds[1] - sbounds[0]) * SAMP;
    if (n > MAXS) n = MAXS;                    // statistically ~never hit
    if (n == 0) {                              // empty graph -> uniform
        if (tid < KCB) out[b * KCB + tid] = 1.0f / (float)KCB;
        return;
    }
    const float* Xb = X + (size_t)node_start * SAMP * DIM;

    // ---- stage codebook into LDS (padded) + column norms ------------------
    for (int idx = tid; idx < KCB * DIM; idx += NTHREADS)
        cb[(idx >> 6) * CBPAD + (idx & 63)] = codebook[idx];
    __syncthreads();
    if (tid < KCB) {
        float s = 0.f;
        #pragma unroll
        for (int d = 0; d < DIM; ++d) { float v = cb[tid * CBPAD + d]; s += v * v; }
        cn[tid] = s;
    }
    // ---- per-sample squared norms (into xn == fS alias) -------------------
    for (int i = tid; i < n; i += NTHREADS) {
        const float4* p = (const float4*)(Xb + (size_t)i * DIM);
        float s = 0.f;
        #pragma unroll
        for (int q = 0; q < DIM / 4; ++q) {
            float4 v = p[q];
            s += v.x * v.x + v.y * v.y + v.z * v.z + v.w * v.w;
        }
        xn[i] = s;
    }
    if (tid < KCB) gS[tid] = 0.f;              // g' = g/eps = 0
    __syncthreads();

    // ---- C' = -max(|x|^2 + |c|^2 - 2 x.c, 0)/eps via f32 WMMA -------------
    // A frag (16x4 f32): lane m16 holds row m16, dims d0+2*kh .. d0+2*kh+1
    // B frag (4x16 f32): lane m16 holds col m16 of codebook, same dims
    // D (16x16 f32, 8 VGPRs): element (M = v + 8*kh, N = m16)
    {
        const int m16 = lane & 15;
        const int kh  = lane >> 4;
        const int ntiles = (n + 15) >> 4;
        for (int t = wv; t < ntiles; t += NWAVES) {
            const int row0 = t << 4;
            int arow = row0 + m16; if (arow >= n) arow = n - 1;
            const float* ap  = Xb + (size_t)arow * DIM + 2 * kh;
            const float* bp0 = cb + m16 * CBPAD + 2 * kh;
            const float* bp1 = cb + (m16 + 16) * CBPAD + 2 * kh;
            v8f acc0 = {0.f,0.f,0.f,0.f,0.f,0.f,0.f,0.f};
            v8f acc1 = {0.f,0.f,0.f,0.f,0.f,0.f,0.f,0.f};
            #pragma unroll
            for (int kk = 0; kk < DIM / 4; ++kk) {
                v2f a  = *(const v2f*)(ap  + kk * 4);
                v2f b0 = *(const v2f*)(bp0 + kk * 4);
                v2f b1 = *(const v2f*)(bp1 + kk * 4);
                acc0 = __builtin_amdgcn_wmma_f32_16x16x4_f32(false, a, false, b0,
                                                             (short)0, acc0, false, false);
                acc1 = __builtin_amdgcn_wmma_f32_16x16x4_f32(false, a, false, b1,
                                                             (short)0, acc1, false, false);
            }
            #pragma unroll
            for (int v = 0; v < 8; ++v) {
                const int r = row0 + v + 8 * kh;
                if (r < n) {
                    float c0 = xn[r] + cn[m16]      - 2.f * acc0[v];
                    float c1 = xn[r] + cn[m16 + 16] - 2.f * acc1[v];
                    // store -C/eps with the max(C,0) clamp folded in
                    Cs[r * CPAD + m16]      = fminf(-INV_EPS * c0, 0.f);
                    Cs[r * CPAD + m16 + 16] = fminf(-INV_EPS * c1, 0.f);
                }
            }
        }
    }
    __syncthreads();

    const float log_a = -__logf((float)n);     // a_i = 1/n_b

    // ---- 50 log-domain Sinkhorn iterations, all in LDS --------------------
    for (int it = 0; it <= ITERS; ++it) {
        // f-step: f'_i = -(logsumexp_k (g'_k + C'_ik) - log K)
        float greg[KCB];
        #pragma unroll
        for (int k = 0; k < KCB; ++k) greg[k] = gS[k];
        for (int i = tid; i < n; i += NTHREADS) {
            const float4* c4 = (const float4*)(Cs + i * CPAD);
            float w[KCB];
            float m = -3.0e38f;
            #pragma unroll
            for (int q = 0; q < KCB / 4; ++q) {
                float4 v = c4[q];
                w[4*q+0] = greg[4*q+0] + v.x;
                w[4*q+1] = greg[4*q+1] + v.y;
                w[4*q+2] = greg[4*q+2] + v.z;
                w[4*q+3] = greg[4*q+3] + v.w;
                m = fmaxf(m, fmaxf(fmaxf(w[4*q+0], w[4*q+1]), fmaxf(w[4*q+2], w[4*q+3])));
            }
            float s = 0.f;
            #pragma unroll
            for (int k = 0; k < KCB; ++k) s += __expf(w[k] - m);
            fS[i] = -(m + __logf(s) - LOG_K);
        }
        __syncthreads();
        if (it == ITERS) break;                // final f computed; skip g

        // g-step: g'_k = -(log_a + seg_logsumexp_i (f'_i + C'_ik))
        {
            float m = -3.0e38f;
            for (int i = wv; i < n; i += NWAVES)
                m = fmaxf(m, fS[i] + Cs[i * CPAD + lane]);
            red[wv * KCB + lane] = m;
            __syncthreads();
            float mk = red[lane];
            #pragma unroll
            for (int j = 1; j < NWAVES; ++j) mk = fmaxf(mk, red[j * KCB + lane]);
            float s = 0.f;
            for (int i = wv; i < n; i += NWAVES)
                s += __expf(fS[i] + Cs[i * CPAD + lane] - mk);
            red2[wv * KCB + lane] = s;
            __syncthreads();
            if (tid < KCB) {
                float S = 0.f;
                #pragma unroll
                for (int j = 0; j < NWAVES; ++j) S += red2[j * KCB + tid];
                gS[tid] = -(log_a + mk + __logf(S));
            }
            __syncthreads();
        }
    }

    // ---- histogram: hist_k = exp(log_a + log_b + seg_lse_i (f'+g'_k+C')) ---
    {
        const float gk = gS[lane];
        float m = -3.0e38f;
        for (int i = wv; i < n; i += NWAVES)
            m = fmaxf(m, fS[i] + gk + Cs[i * CPAD + lane]);
        red[wv * KCB + lane] = m;
        __syncthreads();
        float mk = red[lane];
        #pragma unroll
        for (int j = 1; j < NWAVES; ++j) mk = fmaxf(mk, red[j * KCB + lane]);
        float s = 0.f;
        for (int i = wv; i < n; i += NWAVES)
            s += __expf(fS[i] + gk + Cs[i * CPAD + lane] - mk);
        red2[wv * KCB + lane] = s;
        __syncthreads();
        if (tid < KCB) {
            float S = 0.f;
            #pragma unroll
            for (int j = 0; j < NWAVES; ++j) S += red2[j * KCB + tid];
            red[tid] = __expf(log_a - LOG_K + mk + __logf(S));   // unnormalized hist
        }
        __syncthreads();
        if (tid < KCB) {
            float tot = 0.f;
            #pragma unroll
            for (int j = 0; j < KCB; ++j) tot += red[j];
            out[b * KCB + tid] = red[tid] / (tot + 1e-12f);
        }
    }
}

extern "C" void kernel_launch(void* const* d_in, const int* in_sizes, int n_in,
                              void* d_out, int out_size, void* d_ws, size_t ws_size,
                              hipStream_t stream) {
    const float* X        = (const float*)d_in[0];   // node_distributions [N,4,64]
    const float* codebook = (const float*)d_in[1];   // [32,64]
    const int*   batchidx = (const int*)d_in[2];     // [N]
    float*       out      = (float*)d_out;           // [512,32]
    const int Nnodes = in_sizes[2];

    const size_t smem_bytes = (size_t)SMEM_FLOATS * sizeof(float);  // ~158.8 KiB
    hipLaunchKernelGGL(ot_sinkhorn_hist_kernel,
                       dim3(B_GRAPHS), dim3(NTHREADS), smem_bytes, stream,
                       X, codebook, batchidx, out, Nnodes);
}